// GCN_9491877724923
// MI455X (gfx1250) — compile-verified
//
#include <hip/hip_runtime.h>
#include <cstdint>

typedef __attribute__((ext_vector_type(2))) float v2f;
typedef __attribute__((ext_vector_type(8))) float v8f;

#define NN 100000
#define NE 1600000
#define DF 64
#define NTILES ((NN + 15) / 16)   // 6250 exactly

// ---------------- zero workspace (float4 granularity) ----------------
__global__ void gcn_zero4(float4* __restrict__ p, int n4) {
    int i = blockIdx.x * blockDim.x + threadIdx.x;
    if (i < n4) p[i] = make_float4(0.f, 0.f, 0.f, 0.f);
}

// ---------------- degree over destinations ----------------
__global__ void gcn_degree(const long long* __restrict__ dst,
                           const float* __restrict__ ew,
                           float* __restrict__ deg) {
    int e = blockIdx.x * blockDim.x + threadIdx.x;
    if (e < NE) {
        int d = (int)dst[e];
        __hip_atomic_fetch_add(&deg[d], ew[e], __ATOMIC_RELAXED, __HIP_MEMORY_SCOPE_AGENT);
    }
}

// ---------------- dinv = rsqrt(deg + 1) in place (self-loop weight 1) ----------------
__global__ void gcn_dinv(float* __restrict__ deg) {
    int i = blockIdx.x * blockDim.x + threadIdx.x;
    if (i < NN) deg[i] = rsqrtf(deg[i] + 1.0f);
}

// ---------------- fused edge + self-loop scatter: agg[d] += x[s] * norm ----------------
// 64 consecutive threads handle one edge -> coalesced 256B row gather + row atomics.
__global__ void gcn_scatter(const long long* __restrict__ src,
                            const long long* __restrict__ dst,
                            const float* __restrict__ ew,
                            const float* __restrict__ x,
                            const float* __restrict__ dinv,
                            float* __restrict__ agg) {
    int idx = blockIdx.x * blockDim.x + threadIdx.x;   // < (NE+NN)*64 = 108.8M, fits int32
    int e = idx >> 6;
    int f = idx & 63;
    if (e >= NE + NN) return;
    int s, d;
    float w;
    if (e < NE) {
        s = (int)src[e];
        d = (int)dst[e];
        w = ew[e];
    } else {
        s = d = e - NE;   // self loop, weight 1
        w = 1.0f;
    }
    float nrm = dinv[s] * w * dinv[d];
    float v = x[s * DF + f] * nrm;
    __hip_atomic_fetch_add(&agg[d * DF + f], v, __ATOMIC_RELAXED, __HIP_MEMORY_SCOPE_AGENT);
}

// ---------------- WMMA fp32 GEMM: out = prelu(agg @ W^T + bias) ----------------
// One wave per 16-node tile; W staged in LDS with row stride 66 (bank-conflict-free).
// A(16x4) fragment (ISA 7.12.2): lane L<16 -> row M=L, K={k0,k0+1}; lane L>=16 -> M=L-16, K={k0+2,k0+3}.
// B(4x16) mirrors the K-interleave with N = lane&15.
// C/D(16x16): VGPR r -> M=r (lanes 0-15) / M=r+8 (lanes 16-31), N = lane&15.
__global__ __launch_bounds__(256) void gcn_gemm_prelu(const float* __restrict__ agg,
                                                      const float* __restrict__ W,
                                                      const float* __restrict__ bias,
                                                      const float* __restrict__ prelu_w,
                                                      float* __restrict__ out) {
    __shared__ float Wlds[DF * 66];
    for (int i = threadIdx.x; i < DF * DF; i += 256) {
        int r = i >> 6, c = i & 63;
        Wlds[r * 66 + c] = W[i];          // W[o][k], row-major [64,64]
    }
    __syncthreads();

    int wave = threadIdx.x >> 5;
    int lane = threadIdx.x & 31;
    int tile = blockIdx.x * 8 + wave;
    if (tile >= NTILES) return;           // wave-uniform exit: EXEC stays all-ones for WMMA

    int base = tile * 16;
    int row  = lane & 15;                 // M for A, N-within-tile for B/C
    int half = lane >> 4;
    int koff = half * 2;

    float pw = prelu_w[0];

    v8f acc[4];
#pragma unroll
    for (int nt = 0; nt < 4; ++nt) acc[nt] = (v8f){};

    const float* arow = agg + (base + row) * DF + koff;

#pragma unroll
    for (int ks = 0; ks < 16; ++ks) {
        int k0 = ks * 4;
        v2f a = *(const v2f*)(arow + k0);
#pragma unroll
        for (int nt = 0; nt < 4; ++nt) {
            v2f b = *(const v2f*)(&Wlds[(nt * 16 + row) * 66 + k0 + koff]);
            acc[nt] = __builtin_amdgcn_wmma_f32_16x16x4_f32(
                false, a, false, b, (short)0, acc[nt], false, false);
        }
    }

    // fused bias + PReLU epilogue
#pragma unroll
    for (int nt = 0; nt < 4; ++nt) {
        int n = nt * 16 + row;
        float bn = bias[n];
#pragma unroll
        for (int r = 0; r < 8; ++r) {
            int m = r + half * 8;
            float v = acc[nt][r] + bn;
            out[(base + m) * DF + n] = v > 0.f ? v : pw * v;
        }
    }
}

extern "C" void kernel_launch(void* const* d_in, const int* in_sizes, int n_in,
                              void* d_out, int out_size, void* d_ws, size_t ws_size,
                              hipStream_t stream) {
    const float*     x    = (const float*)d_in[0];
    const long long* eidx = (const long long*)d_in[1];   // int64 [2, NE]
    const float*     ew   = (const float*)d_in[2];
    const float*     W    = (const float*)d_in[3];
    const float*     bias = (const float*)d_in[4];
    const float*     prw  = (const float*)d_in[5];
    float*           out  = (float*)d_out;

    float* deg = (float*)d_ws;            // NN floats
    float* agg = deg + NN;                // NN*64 floats (offset 400000 B, 16B aligned)

    const long long* src = eidx;
    const long long* dst = eidx + NE;

    // 1) zero deg + agg  (NN*65 = 6,500,000 floats = 1,625,000 float4)
    int n4 = (NN * (DF + 1)) / 4;
    gcn_zero4<<<(n4 + 255) / 256, 256, 0, stream>>>((float4*)d_ws, n4);

    // 2) weighted in-degree
    gcn_degree<<<(NE + 255) / 256, 256, 0, stream>>>(dst, ew, deg);

    // 3) dinv (self-loop folded in)
    gcn_dinv<<<(NN + 255) / 256, 256, 0, stream>>>(deg);

    // 4) normalized scatter-add of raw features (edges + self-loops)
    long long items = (long long)(NE + NN) * DF;
    int sblocks = (int)((items + 255) / 256);
    gcn_scatter<<<sblocks, 256, 0, stream>>>(src, dst, ew, x, deg, agg);

    // 5) WMMA GEMM + bias + PReLU -> d_out
    gcn_gemm_prelu<<<(NTILES + 7) / 8, 256, 0, stream>>>(agg, W, bias, prw, out);
}